// MoEGate_40742059770493
// MI455X (gfx1250) — compile-verified
//
#include <hip/hip_runtime.h>
#include <hip/hip_bf16.h>
#include <stdint.h>

typedef float v2f __attribute__((ext_vector_type(2)));
typedef float v8f __attribute__((ext_vector_type(8)));
typedef int   v4i __attribute__((vector_size(16)));

#define H_DIM        2048
#define N_EXPERTS    64
#define TOP_K        6
#define N_GROUP      8
#define TOPK_GROUP   3
#define TOK_PER_BLK  128
#define KC           64
#define NCHUNK       (H_DIM / KC)   // 32
#define LDS_STRIDE   68             // 64 + 4 pad: de-conflicts 64-bank LDS b64 reads

#if __has_builtin(__builtin_amdgcn_global_load_async_to_lds_b128) && \
    __has_builtin(__builtin_amdgcn_s_wait_asynccnt)
#define USE_ASYNC_LDS 1
#else
#define USE_ASYNC_LDS 0
#endif

typedef __attribute__((address_space(1))) v4i* gptr_v4i;
typedef __attribute__((address_space(3))) v4i* lptr_v4i;

__device__ __forceinline__ void copy16(const float* g, float* l) {
#if USE_ASYNC_LDS
    // GLOBAL_LOAD_ASYNC_TO_LDS_B128: memory -> LDS direct, tracked by ASYNCcnt.
    // AS1 pointer built via inttoptr (global generic bits == AS1 bits);
    // AS3 pointer via explicit addrspace cast of the shared-mem pointer.
    __builtin_amdgcn_global_load_async_to_lds_b128(
        (gptr_v4i)(uintptr_t)g,
        (lptr_v4i)l,
        /*imm offset=*/0, /*cpol=*/0);
#else
    *(float4*)l = *(const float4*)g;
#endif
}

__device__ __forceinline__ void copies_done() {
#if USE_ASYNC_LDS
    __builtin_amdgcn_s_wait_asynccnt(0);
#endif
}

// Stage one K-chunk: A = 128 tokens x 64 f32, B = 64 experts x 64 f32.
// 256 threads: thread (row0 = tid>>4, c4 = tid&15) copies float4 column c4 of
// rows row0 + 16*it  -> fully coalesced 256B runs per row.
__device__ __forceinline__ void issue_chunk(
    const float* __restrict__ x, const float* __restrict__ w,
    float* sAbuf, float* sBbuf, int tok0, int h0, int tid)
{
    const int row0 = tid >> 4;
    const int c4   = tid & 15;
    #pragma unroll
    for (int it = 0; it < 8; ++it) {
        int row = row0 + 16 * it;
        copy16(x + (size_t)(tok0 + row) * H_DIM + h0 + c4 * 4,
               &sAbuf[row * LDS_STRIDE + c4 * 4]);
    }
    #pragma unroll
    for (int it = 0; it < 4; ++it) {
        int row = row0 + 16 * it;
        copy16(w + (size_t)row * H_DIM + h0 + c4 * 4,
               &sBbuf[row * LDS_STRIDE + c4 * 4]);
    }
}

__global__ __launch_bounds__(256) void moe_gate_wmma_kernel(
    const float* __restrict__ x,   // [T, H]
    const float* __restrict__ w,   // [E, H]
    float* __restrict__ out,       // [T, TOP_K]
    int T)
{
    __shared__ float sA[2][TOK_PER_BLK * LDS_STRIDE];  // 2 x 34 KB (buf0 reused for logits)
    __shared__ float sB[2][N_EXPERTS  * LDS_STRIDE];   // 2 x 17 KB

    const int tid  = threadIdx.x;
    const int lane = tid & 31;
    const int wave = tid >> 5;       // 0..7 -> token tile
    const int ln16 = lane & 15;
    const int lh   = lane >> 4;      // K-half selector for 32-bit WMMA A/B layout
    const int tok0 = blockIdx.x * TOK_PER_BLK;

    v8f acc[4];
    #pragma unroll
    for (int e = 0; e < 4; ++e)
        #pragma unroll
        for (int i = 0; i < 8; ++i) acc[e][i] = 0.0f;

    // Prologue: prefetch chunk 0 into buffer 0
    issue_chunk(x, w, sA[0], sB[0], tok0, 0, tid);

    for (int c = 0; c < NCHUNK; ++c) {
        copies_done();        // my async copies for chunk c have landed in LDS
        __syncthreads();      // everyone's have; compute(c-1) on the other buf is done

        if (c + 1 < NCHUNK)   // stream next chunk while we compute this one
            issue_chunk(x, w, sA[(c + 1) & 1], sB[(c + 1) & 1],
                        tok0, (c + 1) * KC, tid);

        const float* __restrict__ pA = sA[c & 1];
        const float* __restrict__ pB = sB[c & 1];
        const float* aRow = &pA[(wave * 16 + ln16) * LDS_STRIDE];

        #pragma unroll
        for (int kk = 0; kk < KC; kk += 4) {
            const int k0 = kk + 2 * lh;
            // A 16x4 f32: lane (m = l&15, khalf = l>>4) holds K = {k0, k0+1}
            v2f a = *(const v2f*)(aRow + k0);
            #pragma unroll
            for (int e = 0; e < 4; ++e) {
                // B 4x16 f32 (B = W^T): lane (n = l&15) holds W[n][k0], W[n][k0+1]
                v2f b = *(const v2f*)(&pB[(e * 16 + ln16) * LDS_STRIDE + k0]);
                acc[e] = __builtin_amdgcn_wmma_f32_16x16x4_f32(
                    /*neg_a=*/false, a, /*neg_b=*/false, b,
                    /*c_mod=*/(short)0, acc[e],
                    /*reuse_a=*/false, /*reuse_b=*/false);
            }
        }
    }

    __syncthreads();
    // Spill logits to LDS buf0 as [token][expert] (stride LDS_STRIDE, cols 0..63).
    // C/D 16x16 f32 layout: VGPR r -> M = r (+8 for lanes 16..31), N = lane&15.
    float* lg = sA[0];
    #pragma unroll
    for (int e = 0; e < 4; ++e)
        #pragma unroll
        for (int r = 0; r < 8; ++r) {
            int trow = wave * 16 + r + 8 * lh;
            lg[trow * LDS_STRIDE + e * 16 + ln16] = acc[e][r];
        }
    __syncthreads();

    // Epilogue: one thread per token
    if (tid < TOK_PER_BLK) {
        const float* lrow = &lg[tid * LDS_STRIDE];

        // Pass 1: global max + per-group maxes (per_group = 8)
        float gmax[N_GROUP];
        float M = -3.4e38f;
        #pragma unroll
        for (int g = 0; g < N_GROUP; ++g) {
            float gm = lrow[g * 8];
            #pragma unroll
            for (int e2 = 1; e2 < 8; ++e2) gm = fmaxf(gm, lrow[g * 8 + e2]);
            gmax[g] = gm;
            M = fmaxf(M, gm);
        }
        // Pass 2: full softmax denominator (faithful to reference)
        float Z = 0.0f;
        #pragma unroll
        for (int e2 = 0; e2 < N_EXPERTS; ++e2) Z += __expf(lrow[e2] - M);
        const float invZ = 1.0f / Z;

        // Top-3 groups by group max (softmax is monotone: logit rank == score rank)
        unsigned selMask = 0;
        #pragma unroll
        for (int it = 0; it < TOPK_GROUP; ++it) {
            int bg = 0; float bv = -3.4e38f;
            #pragma unroll
            for (int g = 0; g < N_GROUP; ++g) {
                bool better = (((selMask >> g) & 1u) == 0u) && (gmax[g] > bv);
                bv = better ? gmax[g] : bv;
                bg = better ? g : bg;
            }
            selMask |= (1u << bg);
        }

        // Top-6 logits among selected groups via sorted bubble-insertion
        float best[TOP_K];
        #pragma unroll
        for (int i = 0; i < TOP_K; ++i) best[i] = -3.4e38f;
        #pragma unroll
        for (int e2 = 0; e2 < N_EXPERTS; ++e2) {
            bool sel = ((selMask >> (e2 >> 3)) & 1u) != 0u;
            float v = sel ? lrow[e2] : -3.4e38f;
            #pragma unroll
            for (int i = 0; i < TOP_K; ++i) {
                float hi = fmaxf(best[i], v);
                float lo = fminf(best[i], v);
                best[i] = hi;
                v = lo;
            }
        }

        // Softmax values of the selected 6, renormalized (NORM_TOPK_PROB path)
        float s[TOP_K];
        float denom = 1e-20f;
        #pragma unroll
        for (int i = 0; i < TOP_K; ++i) {
            s[i] = __expf(best[i] - M) * invZ;
            denom += s[i];
        }
        const float invD = 1.0f / denom;
        float* orow = out + (size_t)(tok0 + tid) * TOP_K;
        #pragma unroll
        for (int i = 0; i < TOP_K; ++i) orow[i] = s[i] * invD;
    }
}

extern "C" void kernel_launch(void* const* d_in, const int* in_sizes, int n_in,
                              void* d_out, int out_size, void* d_ws, size_t ws_size,
                              hipStream_t stream) {
    const float* x = (const float*)d_in[0];   // hidden_states [B,S,H] f32
    const float* w = (const float*)d_in[1];   // gate kernel [E,H] f32
    float* out = (float*)d_out;               // [T, TOP_K] f32

    const int T = in_sizes[0] / H_DIM;        // 32768 for the given config
    const int nblocks = T / TOK_PER_BLK;      // 256
    moe_gate_wmma_kernel<<<nblocks, 256, 0, stream>>>(x, w, out, T);
}